// AllAtomDecoder_49976239456316
// MI455X (gfx1250) — compile-verified
//
#include <hip/hip_runtime.h>
#include <hip/hip_bf16.h>

// Problem constants (match reference setup)
#define LRES 128        // residues per batch
#define NSC  34         // side-chain atoms per residue
#define DIM  128        // embedding dim
#define NRT  23         // residue types (only first 20 eligible for argmax)
#define RAD2 64.0f      // RADIUS^2

typedef __attribute__((ext_vector_type(2))) float v2f;
typedef __attribute__((ext_vector_type(8))) float v8f;

// ---------------------------------------------------------------------------
// Kernel 1: per-residue prep.
//   res_type = argmax(aa_pred[..., :20])  (types 20..22 forced to -1e9)
//   vm[s]    = valid_atom37_mask[res_type, 3+s] * mask
//   ca       = bb_pred[:, :, 1, :] * mask
//   S[d]     = sum_s ( vm[s]*res_emb[d] + vm[s]^2 * atom_embed[3+s, d] )
// One block (128 threads) per (b, l).
// ---------------------------------------------------------------------------
__global__ void residue_prep_kernel(const float* __restrict__ aa,
                                    const float* __restrict__ remb,
                                    const float* __restrict__ bb,
                                    const float* __restrict__ mask,
                                    const float* __restrict__ tbl,
                                    const float* __restrict__ aemb,
                                    float* __restrict__ S,
                                    float* __restrict__ vm,
                                    float* __restrict__ ca) {
    const int bl = blockIdx.x;       // b*L + l
    const int d  = threadIdx.x;      // 0..127

    __shared__ float svm[NSC];
    __shared__ float ssum;
    __shared__ int   srt;

    if (d == 0) {
        const float* row = aa + (long)bl * NRT;
        int   best = 0;
        float bv   = row[0];
        for (int t = 1; t < 20; ++t) {        // first-max semantics like jnp.argmax
            float v = row[t];
            if (v > bv) { bv = v; best = t; }
        }
        srt = best;
    }
    __syncthreads();

    const float m = mask[bl];
    if (d < NSC) {
        float v = tbl[srt * 37 + 3 + d] * m;
        svm[d] = v;
        vm[(long)bl * NSC + d] = v;
    }
    if (d >= NSC && d < NSC + 3) {           // threads 34..36 write masked CA
        int c = d - NSC;
        ca[(long)bl * 3 + c] = bb[((long)bl * 4 + 1) * 3 + c] * m;
    }
    __syncthreads();

    if (d == 0) {
        float s = 0.f;
        for (int i = 0; i < NSC; ++i) s += svm[i];
        ssum = s;
    }
    __syncthreads();

    float acc = ssum * remb[(long)bl * DIM + d];
    #pragma unroll 2
    for (int s = 0; s < NSC; ++s) {
        float v = svm[s];
        acc = fmaf(v * v, aemb[(3 + s) * DIM + d], acc);
    }
    S[(long)bl * DIM + d] = acc;
}

// ---------------------------------------------------------------------------
// Kernel 2: residue-level adjacency (diagonal included: d2(ri,ri)=0 < R^2).
// Grid (L, B), block L threads: thread rj handles pair (ri=blockIdx.x, rj).
// ---------------------------------------------------------------------------
__global__ void adjacency_kernel(const float* __restrict__ ca,
                                 float* __restrict__ adj) {
    const int b  = blockIdx.y;
    const int ri = blockIdx.x;
    const int rj = threadIdx.x;
    const float* cb = ca + (long)b * LRES * 3;
    float dx = cb[ri * 3 + 0] - cb[rj * 3 + 0];
    float dy = cb[ri * 3 + 1] - cb[rj * 3 + 1];
    float dz = cb[ri * 3 + 2] - cb[rj * 3 + 2];
    float d2 = dx * dx + dy * dy + dz * dz;
    adj[((long)b * LRES + ri) * LRES + rj] = (d2 < RAD2) ? 1.0f : 0.0f;
}

// ---------------------------------------------------------------------------
// Kernel 3: T = AdjRes @ S per batch, exact f32 via V_WMMA_F32_16X16X4_F32.
// One wave32 per 16x16 output tile; K loop over 128 in steps of 4.
//   A (16x4 f32): lane m=lane&15; VGPR0 holds K=koff, VGPR1 K=koff+1 where
//                 koff = (lane>=16 ? 2 : 0)   (ISA 7.12.2 layout)
//   B (4x16 f32): mirrored striping (row K across lanes within a VGPR)
//   C/D (16x16):  VGPR v -> row v + (lane>=16 ? 8 : 0), col = lane&15
// EXEC is all-ones (flat 32-thread block, no divergence) as WMMA requires.
// ---------------------------------------------------------------------------
__global__ void adj_matmul_kernel(const float* __restrict__ adj,
                                  const float* __restrict__ S,
                                  float* __restrict__ T) {
    const int b    = blockIdx.y;
    const int mt   = blockIdx.x / (DIM / 16);
    const int nt   = blockIdx.x % (DIM / 16);
    const int lane = threadIdx.x;       // 0..31
    const int half = lane >> 4;         // 0 or 1
    const int lm   = lane & 15;

    const float* A  = adj + (long)b * LRES * LRES;
    const float* Bm = S   + (long)b * LRES * DIM;

    const int m = mt * 16 + lm;         // output row handled by this lane (A side)
    const int n = nt * 16 + lm;         // output col handled by this lane (B/C side)

    v8f c = {};
    for (int k0 = 0; k0 < LRES; k0 += 4) {
        const int ka = k0 + half * 2;
        v2f av, bv;
        av.x = A[(long)m * LRES + ka];
        av.y = A[(long)m * LRES + ka + 1];
        bv.x = Bm[(long)ka * DIM + n];
        bv.y = Bm[(long)(ka + 1) * DIM + n];
        c = __builtin_amdgcn_wmma_f32_16x16x4_f32(
                /*neg_a=*/false, av, /*neg_b=*/false, bv,
                /*c_mod=*/(short)0, c, /*reuse_a=*/false, /*reuse_b=*/false);
    }

    #pragma unroll
    for (int v = 0; v < 8; ++v) {
        const int row = mt * 16 + v + half * 8;
        T[((long)b * LRES + row) * DIM + n] = c[v];
    }
}

// ---------------------------------------------------------------------------
// Kernel 4: expand residue-level T to atom-level output with self removal.
//   msg[b, l*34+s, d] = (vm>0) ? T[b,l,d] - (vm*res_emb[d] + vm^2*atom_emb)
//                              : 0
// One block (128 threads) per (b, l); writes 34 coalesced rows of 128.
// ---------------------------------------------------------------------------
__global__ void expand_kernel(const float* __restrict__ T,
                              const float* __restrict__ remb,
                              const float* __restrict__ aemb,
                              const float* __restrict__ vm,
                              float* __restrict__ out) {
    const int bl = blockIdx.x;
    const int d  = threadIdx.x;

    __shared__ float svm[NSC];
    if (d < NSC) svm[d] = vm[(long)bl * NSC + d];
    __syncthreads();

    const float t  = T[(long)bl * DIM + d];
    const float re = remb[(long)bl * DIM + d];

    #pragma unroll 2
    for (int s = 0; s < NSC; ++s) {
        float v  = svm[s];
        float sc = v * re + v * v * aemb[(3 + s) * DIM + d];
        out[((long)bl * NSC + s) * DIM + d] = (v > 0.f) ? (t - sc) : 0.f;
    }
}

// ---------------------------------------------------------------------------
// Launch
// ---------------------------------------------------------------------------
extern "C" void kernel_launch(void* const* d_in, const int* in_sizes, int n_in,
                              void* d_out, int out_size, void* d_ws, size_t ws_size,
                              hipStream_t stream) {
    const float* aa   = (const float*)d_in[0];   // (B, L, 23)
    const float* remb = (const float*)d_in[1];   // (B, L, 128)
    const float* bb   = (const float*)d_in[2];   // (B, L, 4, 3)
    const float* mask = (const float*)d_in[3];   // (B, L)
    const float* tbl  = (const float*)d_in[4];   // (23, 37)
    const float* aemb = (const float*)d_in[5];   // (37, 128)
    float* out = (float*)d_out;                  // (B, L*34, 128)

    const int BL = in_sizes[3];                  // B * L
    const int B  = BL / LRES;

    // Workspace layout (floats):
    float* ws    = (float*)d_ws;
    float* S     = ws;                            // BL * DIM
    float* T     = S  + (size_t)BL * DIM;         // BL * DIM
    float* adj   = T  + (size_t)BL * DIM;         // B * L * L
    float* vm    = adj + (size_t)B * LRES * LRES; // BL * NSC
    float* ca    = vm + (size_t)BL * NSC;         // BL * 3
    (void)ws_size; (void)n_in; (void)out_size;

    residue_prep_kernel<<<BL, DIM, 0, stream>>>(aa, remb, bb, mask, tbl, aemb,
                                                S, vm, ca);

    adjacency_kernel<<<dim3(LRES, B), LRES, 0, stream>>>(ca, adj);

    adj_matmul_kernel<<<dim3((LRES / 16) * (DIM / 16), B), 32, 0, stream>>>(adj, S, T);

    expand_kernel<<<BL, DIM, 0, stream>>>(T, remb, aemb, vm, out);
}